// BM2_15822659518813
// MI455X (gfx1250) — compile-verified
//
#include <hip/hip_runtime.h>
#include <hip/hip_bf16.h>
#include <math.h>

typedef __attribute__((ext_vector_type(2))) float v2f;
typedef __attribute__((ext_vector_type(8))) float v8f;

#define TILE_K  32
#define TILE_N  64
#define PAIRS   (TILE_K / 2)   // 16 K-pair rows in LDS tile
#define PSTRIDE 160            // floats per pair-row: 128 data + 32 pad
                               // -> pairRow & pairRow+1 are 32 banks apart (conflict-free b64 reads)

// ---------------------------------------------------------------------------
// Per-(b,c) mean over HW
__global__ __launch_bounds__(256)
void k_reduce_mean(const float* __restrict__ x, float* __restrict__ outMean, int HW) {
    __shared__ float sm[256];
    long base = (long)blockIdx.x * HW;
    float s = 0.f;
    for (int i = threadIdx.x; i < HW; i += 256) s += x[base + i];
    sm[threadIdx.x] = s; __syncthreads();
    for (int st = 128; st > 0; st >>= 1) {
        if (threadIdx.x < st) sm[threadIdx.x] += sm[threadIdx.x + st];
        __syncthreads();
    }
    if (threadIdx.x == 0) outMean[blockIdx.x] = sm[0] / (float)HW;
}

// Per-(b,c) mean AND max over HW (channel attention descriptors)
__global__ __launch_bounds__(256)
void k_reduce_mean_max(const float* __restrict__ x, float* __restrict__ outMean,
                       float* __restrict__ outMax, int HW) {
    __shared__ float sm[256];
    __shared__ float mm[256];
    long base = (long)blockIdx.x * HW;
    float s = 0.f, m = -3.402823466e38f;
    for (int i = threadIdx.x; i < HW; i += 256) {
        float v = x[base + i];
        s += v; m = fmaxf(m, v);
    }
    sm[threadIdx.x] = s; mm[threadIdx.x] = m; __syncthreads();
    for (int st = 128; st > 0; st >>= 1) {
        if (threadIdx.x < st) {
            sm[threadIdx.x] += sm[threadIdx.x + st];
            mm[threadIdx.x] = fmaxf(mm[threadIdx.x], mm[threadIdx.x + st]);
        }
        __syncthreads();
    }
    if (threadIdx.x == 0) {
        outMean[blockIdx.x] = sm[0] / (float)HW;
        outMax[blockIdx.x]  = mm[0];
    }
}

// ---------------------------------------------------------------------------
// Dynamic-conv attention head: relu(fc1 @ pooled) -> fc2 -> softmax(/34)
__global__ __launch_bounds__(256)
void k_dyn_att(const float* __restrict__ pooled, const float* __restrict__ fc1,
               const float* __restrict__ fc2, const float* __restrict__ fc2b,
               float* __restrict__ att, int Ci, int Hdim) {
    __shared__ float h[264];
    __shared__ float lg[4];
    const int b = blockIdx.x, t = threadIdx.x;
    const float* p = pooled + (long)b * Ci;
    for (int hid = t; hid < Hdim; hid += 256) {
        float s = 0.f;
        for (int c = 0; c < Ci; ++c) s += fc1[(long)hid * Ci + c] * p[c];
        h[hid] = s > 0.f ? s : 0.f;
    }
    __syncthreads();
    if (t < 4) {
        float s = fc2b[t];
        for (int j = 0; j < Hdim; ++j) s += fc2[(long)t * Hdim + j] * h[j];
        lg[t] = s * (1.0f / 34.0f);   // softmax temperature
    }
    __syncthreads();
    if (t == 0) {
        float mx = fmaxf(fmaxf(lg[0], lg[1]), fmaxf(lg[2], lg[3]));
        float e0 = expf(lg[0] - mx), e1 = expf(lg[1] - mx);
        float e2 = expf(lg[2] - mx), e3 = expf(lg[3] - mx);
        float inv = 1.f / (e0 + e1 + e2 + e3);
        att[b * 4 + 0] = e0 * inv; att[b * 4 + 1] = e1 * inv;
        att[b * 4 + 2] = e2 * inv; att[b * 4 + 3] = e3 * inv;
    }
}

// Mix K=4 expert kernels with attention; optional per-input-channel mask fold.
__global__ __launch_bounds__(256)
void k_combine_aw(const float* __restrict__ w, const float* __restrict__ bias,
                  const float* __restrict__ att, const float* __restrict__ mask,
                  float* __restrict__ aw, float* __restrict__ ab, int Co, int Ci) {
    const int b = blockIdx.z;
    const int e = blockIdx.x * 256 + threadIdx.x;
    const int tot = Co * Ci;
    const float a0 = att[b*4+0], a1 = att[b*4+1], a2 = att[b*4+2], a3 = att[b*4+3];
    if (e < tot) {
        float v = a0 * w[e] + a1 * w[(long)tot + e] + a2 * w[2L*tot + e] + a3 * w[3L*tot + e];
        if (mask) v *= mask[(long)b * Ci + (e % Ci)];
        aw[(long)b * tot + e] = v;
    }
    if (e < Co) {
        ab[(long)b * Co + e] = a0*bias[e] + a1*bias[Co+e] + a2*bias[2*Co+e] + a3*bias[3*Co+e];
    }
}

// Channel-attention MLP: mask = sigmoid(w2@(relu_h(avg)+relu_h(max)) + 2*b2)
// Also emits pooled3 = mask * avg  (== mean(lr*mask), pooling for dl1 attention).
__global__ __launch_bounds__(256)
void k_ca_mlp(const float* __restrict__ avg, const float* __restrict__ mx,
              const float* __restrict__ w1, const float* __restrict__ b1,
              const float* __restrict__ w2, const float* __restrict__ b2,
              float* __restrict__ mask, float* __restrict__ pooled3) {
    __shared__ float s[128];
    const int b = blockIdx.x, t = threadIdx.x;
    const float* av = avg + (long)b * 256;
    const float* mv = mx  + (long)b * 256;
    if (t < 128) {
        float ha = b1[t], hm = b1[t];
        for (int c = 0; c < 256; ++c) {
            float w = w1[(long)t * 256 + c];
            ha += w * av[c]; hm += w * mv[c];
        }
        ha = ha > 0.f ? ha : 0.f;
        hm = hm > 0.f ? hm : 0.f;
        s[t] = ha + hm;
    }
    __syncthreads();
    float acc = 2.0f * b2[t];
    for (int j = 0; j < 128; ++j) acc += s[j] * w2[(long)t * 128 + j];
    float m = 1.0f / (1.0f + expf(-acc));
    mask[(long)b * 256 + t]    = m;
    pooled3[(long)b * 256 + t] = m * av[t];
}

// ---------------------------------------------------------------------------
// Per-sample GEMM via V_WMMA_F32_16X16X4_F32:
//   Y[b] (M=128 x N) = AW[b] (128 x Kdim) @ X[b] (Kdim x N) + AB[b]
// Block = 256 threads (8 waves); wave w owns rows [16w,16w+16), N-strip of 64.
// X tile (32K x 64N) staged via b128 loads into a WMMA-fragment-ready LDS
// layout: tile[(pairRow)*PSTRIDE + 2*n + elem] = X[k + 2*pairRow + elem][n0+n],
// so every B fragment is a single aligned ds_load_b64 (no register shuffles).
// upsample=1: N=1024 is a 32x32 image, each value scattered 2x2 into 64x64.
__global__ __launch_bounds__(256)
void k_dyn_gemm_wmma(const float* __restrict__ X, const float* __restrict__ AW,
                     const float* __restrict__ AB, float* __restrict__ Y,
                     int Kdim, int N, long yStrideB, int upsample, int yChanOffset) {
    __shared__ float tileX[PAIRS * PSTRIDE];
    const int b    = blockIdx.z;
    const int t    = threadIdx.x;
    const int lane = t & 31;
    const int wv   = t >> 5;        // 0..7
    const int hi   = lane >> 4;     // 0/1 half-wave
    const int lo   = lane & 15;
    const int n0   = blockIdx.x * TILE_N;
    const int m0   = wv * 16;

    const float* Xb = X  + (long)b * Kdim * N;
    const float* Ab = AW + (long)b * 128 * Kdim;
    const float* Bb = AB + (long)b * 128;

    v8f acc[4];
    acc[0] = (v8f){}; acc[1] = (v8f){}; acc[2] = (v8f){}; acc[3] = (v8f){};

    for (int k = 0; k < Kdim; k += TILE_K) {
        __syncthreads();
        // stage: 512 float4s, 2 per thread, coalesced b128 global loads
        #pragma unroll
        for (int i = 0; i < 2; ++i) {
            int idx = t + 256 * i;          // 0..511
            int r   = idx >> 4;             // K row within tile: 0..31
            int c0  = (idx & 15) << 2;      // N col: 0,4,...,60
            const float4 v = *(const float4*)(Xb + (long)(k + r) * N + n0 + c0);
            int pr   = ((r >> 2) << 1) + ((r >> 1) & 1);  // pair-row
            int elem = r & 1;
            float* dst = &tileX[pr * PSTRIDE + 2 * c0 + elem];
            dst[0] = v.x; dst[2] = v.y; dst[4] = v.z; dst[6] = v.w;
        }
        __syncthreads();
        #pragma unroll
        for (int kk = 0; kk < TILE_K; kk += 4) {
            // A fragment: lane<16 -> K=k+kk,k+kk+1 ; lane>=16 -> K=k+kk+2,k+kk+3
            v2f a = *(const v2f*)(Ab + (long)(m0 + lo) * Kdim + (k + kk + 2 * hi));
            const float* tb = &tileX[(kk / 2 + hi) * PSTRIDE + 2 * lo];
            #pragma unroll
            for (int j = 0; j < 4; ++j) {
                v2f bf = *(const v2f*)(tb + 32 * j);   // single ds_load_b64
                acc[j] = __builtin_amdgcn_wmma_f32_16x16x4_f32(
                    false, a, false, bf, (short)0, acc[j], false, false);
            }
        }
    }

    float* Yb = Y + (long)b * yStrideB;
    #pragma unroll
    for (int j = 0; j < 4; ++j) {
        int n = n0 + 16 * j + lo;
        #pragma unroll
        for (int r = 0; r < 8; ++r) {
            int m = m0 + r + 8 * hi;               // C/D layout: VGPR r -> M=r (+8 upper half)
            float val = acc[j][r] + Bb[m];
            int ch = yChanOffset + m;
            if (!upsample) {
                Yb[(long)ch * N + n] = val;
            } else {
                int h = n >> 5, w = n & 31;        // 32x32 source pixel
                float* p = Yb + (long)ch * 4096 + (h * 2) * 64 + (w * 2);
                v2f dup; dup.x = val; dup.y = val;
                *(v2f*)p        = dup;             // b64 store: (2h, 2w..2w+1)
                *(v2f*)(p + 64) = dup;             // b64 store: (2h+1, 2w..2w+1)
            }
        }
    }
}

// ---------------------------------------------------------------------------
extern "C" void kernel_launch(void* const* d_in, const int* in_sizes, int n_in,
                              void* d_out, int out_size, void* d_ws, size_t ws_size,
                              hipStream_t stream) {
    const int B = 32, C1 = 128, C2 = 256;
    const int HW2 = 64 * 64, HW3 = 32 * 32;

    const float* x2        = (const float*)d_in[0];
    const float* x3        = (const float*)d_in[1];
    const float* u1_w      = (const float*)d_in[2];
    const float* u1_b      = (const float*)d_in[3];
    const float* u1_fc1_w  = (const float*)d_in[4];
    const float* u1_fc2_w  = (const float*)d_in[5];
    const float* u1_fc2_b  = (const float*)d_in[6];
    const float* u2_w      = (const float*)d_in[7];
    const float* u2_b      = (const float*)d_in[8];
    const float* u2_fc1_w  = (const float*)d_in[9];
    const float* u2_fc2_w  = (const float*)d_in[10];
    const float* u2_fc2_b  = (const float*)d_in[11];
    const float* dl1_w     = (const float*)d_in[12];
    const float* dl1_b     = (const float*)d_in[13];
    const float* dl1_fc1_w = (const float*)d_in[14];
    const float* dl1_fc2_w = (const float*)d_in[15];
    const float* dl1_fc2_b = (const float*)d_in[16];
    const float* ca_w1     = (const float*)d_in[17];
    const float* ca_b1     = (const float*)d_in[18];
    const float* ca_w2     = (const float*)d_in[19];
    const float* ca_b2     = (const float*)d_in[20];
    float* out = (float*)d_out;

    // workspace layout (floats)
    float* ws = (float*)d_ws;
    long off = 0;
    float* lr      = ws + off; off += (long)B * 2 * C1 * HW2;   // 33,554,432
    float* aw2     = ws + off; off += (long)B * C1 * C1;
    float* ab2     = ws + off; off += (long)B * C1;
    float* aw1     = ws + off; off += (long)B * C1 * C2;
    float* ab1     = ws + off; off += (long)B * C1;
    float* aw3     = ws + off; off += (long)B * C1 * C2;
    float* ab3     = ws + off; off += (long)B * C1;
    float* pool2   = ws + off; off += (long)B * C1;
    float* pool3   = ws + off; off += (long)B * C2;
    float* att2    = ws + off; off += (long)B * 4;
    float* att1    = ws + off; off += (long)B * 4;
    float* att3    = ws + off; off += (long)B * 4;
    float* ca_avg  = ws + off; off += (long)B * 2 * C1;
    float* ca_max  = ws + off; off += (long)B * 2 * C1;
    float* maskBuf = ws + off; off += (long)B * 2 * C1;
    float* pooled3 = ws + off; off += (long)B * 2 * C1;

    // 1) attention pooling of raw inputs
    k_reduce_mean<<<B * C1, 256, 0, stream>>>(x2, pool2, HW2);
    k_reduce_mean<<<B * C2, 256, 0, stream>>>(x3, pool3, HW3);

    // 2) attention softmax heads
    k_dyn_att<<<B, 256, 0, stream>>>(pool2, u2_fc1_w, u2_fc2_w, u2_fc2_b, att2, C1, C1 + 1);
    k_dyn_att<<<B, 256, 0, stream>>>(pool3, u1_fc1_w, u1_fc2_w, u1_fc2_b, att1, C2, C2 + 1);

    // 3) mix expert kernels
    k_combine_aw<<<dim3((C1 * C1 + 255) / 256, 1, B), 256, 0, stream>>>(
        u2_w, u2_b, att2, nullptr, aw2, ab2, C1, C1);
    k_combine_aw<<<dim3((C1 * C2 + 255) / 256, 1, B), 256, 0, stream>>>(
        u1_w, u1_b, att1, nullptr, aw1, ab1, C1, C2);

    const long lrStrideB = (long)2 * C1 * HW2;

    // 4) u2 GEMM -> lr channels [0,128)
    k_dyn_gemm_wmma<<<dim3(HW2 / TILE_N, 1, B), 256, 0, stream>>>(
        x2, aw2, ab2, lr, C1, HW2, lrStrideB, /*upsample=*/0, /*chanOff=*/0);

    // 5) u1 GEMM + 2x nearest upsample -> lr channels [128,256)
    k_dyn_gemm_wmma<<<dim3(HW3 / TILE_N, 1, B), 256, 0, stream>>>(
        x3, aw1, ab1, lr, C2, HW3, lrStrideB, /*upsample=*/1, /*chanOff=*/C1);

    // 6) channel-attention descriptors over lr
    k_reduce_mean_max<<<B * 2 * C1, 256, 0, stream>>>(lr, ca_avg, ca_max, HW2);

    // 7) CA MLP -> mask, and pooled3 = mask * avg (dl1 attention pooling, fused)
    k_ca_mlp<<<B, 256, 0, stream>>>(ca_avg, ca_max, ca_w1, ca_b1, ca_w2, ca_b2,
                                    maskBuf, pooled3);

    // 8) dl1 attention head
    k_dyn_att<<<B, 256, 0, stream>>>(pooled3, dl1_fc1_w, dl1_fc2_w, dl1_fc2_b,
                                     att3, 2 * C1, 2 * C1 + 1);

    // 9) mix dl1 experts with mask folded into the K-columns
    k_combine_aw<<<dim3((C1 * C2 + 255) / 256, 1, B), 256, 0, stream>>>(
        dl1_w, dl1_b, att3, maskBuf, aw3, ab3, C1, C2);

    // 10) dl1 GEMM -> output
    k_dyn_gemm_wmma<<<dim3(HW2 / TILE_N, 1, B), 256, 0, stream>>>(
        lr, aw3, ab3, out, C2, HW2, (long)C1 * HW2, /*upsample=*/0, /*chanOff=*/0);
}